// GraphSAGE_14946486190200
// MI455X (gfx1250) — compile-verified
//
#include <hip/hip_runtime.h>

#define FD 64  // feature dim

typedef __attribute__((ext_vector_type(2))) float v2f;
typedef __attribute__((ext_vector_type(8))) float v8f;

// ---------------------------------------------------------------------------
// summ[n][d] = x[n][d]  (self-loop term); optionally deg[n] = 1.0
// ---------------------------------------------------------------------------
__global__ void sage_init_kernel(float* __restrict__ summ,
                                 const float* __restrict__ x,
                                 float* __restrict__ deg,
                                 int n_nodes, int init_deg) {
  int i = blockIdx.x * blockDim.x + threadIdx.x;
  int total = n_nodes * FD;
  if (i < total) summ[i] = x[i];
  if (init_deg && i < n_nodes) deg[i] = 1.0f;
}

// ---------------------------------------------------------------------------
// deg[dst[e]] += 1  (in-degree; same for both layers)
// ---------------------------------------------------------------------------
__global__ void sage_deg_kernel(float* __restrict__ deg,
                                const int* __restrict__ dst, int n_edges) {
  int e = blockIdx.x * blockDim.x + threadIdx.x;
  if (e < n_edges) unsafeAtomicAdd(&deg[dst[e]], 1.0f);
}

// ---------------------------------------------------------------------------
// summ[dst[e]][:] += x[src[e]][:]   16 threads per edge, float4 per thread.
// Gather + atomic scatter both land in L2 (feature tensor is 12.8 MB << 192 MB).
// ---------------------------------------------------------------------------
__global__ void sage_scatter_kernel(float* __restrict__ summ,
                                    const float* __restrict__ x,
                                    const int* __restrict__ src,
                                    const int* __restrict__ dst,
                                    int n_edges) {
  int t = blockIdx.x * blockDim.x + threadIdx.x;
  int e = t >> 4;
  if (e >= n_edges) return;
  int f = (t & 15) * 4;
  const float4 v = *(const float4*)(x + (size_t)src[e] * FD + f);
  float* o = summ + (size_t)dst[e] * FD + f;
  unsafeAtomicAdd(o + 0, v.x);
  unsafeAtomicAdd(o + 1, v.y);
  unsafeAtomicAdd(o + 2, v.z);
  unsafeAtomicAdd(o + 3, v.w);
}

// ---------------------------------------------------------------------------
// out = relu( (summ/deg) @ Wl + bl + x @ Wr )
// One block = 16 nodes, 4 wave32s; wave w computes output columns [16w,16w+16).
// K=64 accumulated with chained v_wmma_f32_16x16x4_f32 (full f32 precision).
//
// ISA fragment layouts (f32, wave32):
//   A 16x4:  lane L holds row M=L%16; VGPR v = K = kbase + v + 2*(L/16)... for
//            the 16x4 table: lanes 0-15 -> K={0,1}, lanes 16-31 -> K={2,3};
//            i.e. a[v] = A[M][4*ks + 2*half + v]
//   B 4x16:  row K striped across lanes within a VGPR:
//            b[v] = B[4*ks + 2*half + v][N = lane%16]
//   C 16x16: c[v] -> row M = v + 8*(lane/16), col N = lane%16
// ---------------------------------------------------------------------------
__global__ __launch_bounds__(128) void sage_gemm_kernel(
    const float* __restrict__ summ, const float* __restrict__ deg,
    const float* __restrict__ xin,
    const float* __restrict__ Wl, const float* __restrict__ bl,
    const float* __restrict__ Wr,
    float* __restrict__ out, int n_nodes) {
  // +1 column pad: row stride 65 -> 16 rows of one column hit 16 distinct banks
  __shared__ float sMean[16][FD + 1];
  __shared__ float sX[16][FD + 1];

  const int base = blockIdx.x * 16;
  const int tid = threadIdx.x;

  for (int i = tid; i < 16 * FD; i += 128) {
    int m = i >> 6;
    int k = i & (FD - 1);
    int node = base + m;
    if (node < n_nodes) {
      float dg = deg[node];
      sMean[m][k] = summ[(size_t)node * FD + k] / dg;
      sX[m][k]   = xin[(size_t)node * FD + k];
    } else {
      sMean[m][k] = 0.0f;
      sX[m][k]   = 0.0f;
    }
  }
  __syncthreads();

  const int wave = tid >> 5;   // output n-tile 0..3
  const int lane = tid & 31;
  const int half = lane >> 4;  // 0: lanes 0-15, 1: lanes 16-31
  const int l16  = lane & 15;
  const int ncol = wave * 16 + l16;

  v8f acc = {};

  // mean @ Wl
#pragma unroll
  for (int ks = 0; ks < FD / 4; ++ks) {
    const int kb = ks * 4 + 2 * half;
    v2f a, b;
    a[0] = sMean[l16][kb];
    a[1] = sMean[l16][kb + 1];
    b[0] = Wl[(kb)     * FD + ncol];
    b[1] = Wl[(kb + 1) * FD + ncol];
    acc = __builtin_amdgcn_wmma_f32_16x16x4_f32(false, a, false, b,
                                                (short)0, acc, false, false);
  }
  // + x @ Wr  (accumulate into same C)
#pragma unroll
  for (int ks = 0; ks < FD / 4; ++ks) {
    const int kb = ks * 4 + 2 * half;
    v2f a, b;
    a[0] = sX[l16][kb];
    a[1] = sX[l16][kb + 1];
    b[0] = Wr[(kb)     * FD + ncol];
    b[1] = Wr[(kb + 1) * FD + ncol];
    acc = __builtin_amdgcn_wmma_f32_16x16x4_f32(false, a, false, b,
                                                (short)0, acc, false, false);
  }

  const float bias = bl[ncol];
#pragma unroll
  for (int v = 0; v < 8; ++v) {
    int m = v + 8 * half;
    int node = base + m;
    if (node < n_nodes) {
      float val = acc[v] + bias;
      out[(size_t)node * FD + ncol] = val > 0.0f ? val : 0.0f;
    }
  }
}

// ---------------------------------------------------------------------------
extern "C" void kernel_launch(void* const* d_in, const int* in_sizes, int n_in,
                              void* d_out, int out_size, void* d_ws, size_t ws_size,
                              hipStream_t stream) {
  const float* x   = (const float*)d_in[0];
  const int*   ei  = (const int*)d_in[1];
  const float* Wl0 = (const float*)d_in[2];
  const float* bl0 = (const float*)d_in[3];
  const float* Wr0 = (const float*)d_in[4];
  const float* Wl1 = (const float*)d_in[5];
  const float* bl1 = (const float*)d_in[6];
  const float* Wr1 = (const float*)d_in[7];

  const int n_nodes = in_sizes[0] / FD;
  const int n_edges = in_sizes[1] / 2;
  const int* src = ei;             // edge_index[0]
  const int* dst = ei + n_edges;   // edge_index[1]

  float* summ = (float*)d_ws;                       // N*FD
  float* h1   = summ + (size_t)n_nodes * FD;        // N*FD
  float* deg  = h1   + (size_t)n_nodes * FD;        // N
  float* outp = (float*)d_out;

  const int T = 256;
  const int grid_nd = (n_nodes * FD + T - 1) / T;
  const int grid_e  = (n_edges + T - 1) / T;
  const int grid_sc = (n_edges * 16 + T - 1) / T;
  const int grid_g  = (n_nodes + 15) / 16;

  // ---- layer 0 ----
  sage_init_kernel<<<grid_nd, T, 0, stream>>>(summ, x, deg, n_nodes, 1);
  sage_deg_kernel<<<grid_e, T, 0, stream>>>(deg, dst, n_edges);
  sage_scatter_kernel<<<grid_sc, T, 0, stream>>>(summ, x, src, dst, n_edges);
  sage_gemm_kernel<<<grid_g, 128, 0, stream>>>(summ, deg, x, Wl0, bl0, Wr0,
                                               h1, n_nodes);
  // ---- layer 1 (deg unchanged) ----
  sage_init_kernel<<<grid_nd, T, 0, stream>>>(summ, h1, deg, n_nodes, 0);
  sage_scatter_kernel<<<grid_sc, T, 0, stream>>>(summ, h1, src, dst, n_edges);
  sage_gemm_kernel<<<grid_g, 128, 0, stream>>>(summ, deg, h1, Wl1, bl1, Wr1,
                                               outp, n_nodes);
}